// DiscAdvLossForTarget_min_1606317768951
// MI455X (gfx1250) — compile-verified
//
#include <hip/hip_runtime.h>
#include <hip/hip_bf16.h>
#include <stdint.h>

// Problem constants (reference: B=65536, C=1000, input is B x (C+1) f32)
#define N_ROWS      65536
#define N_COLS      1001      // 1000 logits + 1 "extra"
#define N_LOGITS    1000
#define WAVES_PB    8
#define THREADS_PB  256
#define N_BLOCKS    1024      // 8192 waves -> exactly 8 rows per wave
#define ROWS_PW     (N_ROWS / (N_BLOCKS * WAVES_PB))   // = 8
#define SLOT_FLOATS 1024      // per-buffer LDS slot (4 KB), 16B aligned

__device__ __forceinline__ float wave_sum(float v) {
    v += __shfl_xor(v, 16, 32);
    v += __shfl_xor(v,  8, 32);
    v += __shfl_xor(v,  4, 32);
    v += __shfl_xor(v,  2, 32);
    v += __shfl_xor(v,  1, 32);
    return v;
}

__device__ __forceinline__ float wave_max(float v) {
    v = fmaxf(v, __shfl_xor(v, 16, 32));
    v = fmaxf(v, __shfl_xor(v,  8, 32));
    v = fmaxf(v, __shfl_xor(v,  4, 32));
    v = fmaxf(v, __shfl_xor(v,  2, 32));
    v = fmaxf(v, __shfl_xor(v,  1, 32));
    return v;
}

// Issue one row (1001 f32, only 4B-aligned) as 32 async b32 global->LDS copies.
// Streamed-once data: NT temporal hint (input is 262 MB > 192 MB L2).
__device__ __forceinline__ void issue_row_async(const float* __restrict__ rb,
                                                uint32_t slot_lds, int lane)
{
    #pragma unroll 4
    for (int r = 0; r < 31; ++r) {
        const int idx = r * 32 + lane;
        const uint32_t la = slot_lds + (uint32_t)idx * 4u;
        const float* gp = rb + idx;
        asm volatile("global_load_async_to_lds_b32 %0, %1, off th:TH_LOAD_NT"
                     :: "v"(la), "v"(gp) : "memory");
    }
    const int idx = 31 * 32 + lane;           // 992..1023, keep idx <= 1000
    if (idx < N_COLS) {
        const uint32_t la = slot_lds + (uint32_t)idx * 4u;
        const float* gp = rb + idx;
        asm volatile("global_load_async_to_lds_b32 %0, %1, off th:TH_LOAD_NT"
                     :: "v"(la), "v"(gp) : "memory");
    }
}

// One wave per row, double-buffered: while row i is processed out of LDS,
// row i+1's 32 async b32 copies are already in flight (ASYNCcnt-ordered, so
// waiting asynccnt<=32 releases exactly the older row's group).
__global__ __launch_bounds__(THREADS_PB)
void disc_adv_loss_rows(const float* __restrict__ in, float* __restrict__ partial)
{
    __shared__ __align__(16) float smem[WAVES_PB * 2 * SLOT_FLOATS];  // 64 KB
    __shared__ float wacc[WAVES_PB];

    const int lane = threadIdx.x & 31;
    const int wid  = threadIdx.x >> 5;
    float* slot0 = &smem[(wid * 2 + 0) * SLOT_FLOATS];
    float* slot1 = &smem[(wid * 2 + 1) * SLOT_FLOATS];
    // Low 32 bits of a generic pointer into the LDS aperture = LDS byte offset.
    const uint32_t lds0 = (uint32_t)(uintptr_t)slot0;
    const uint32_t lds1 = (uint32_t)(uintptr_t)slot1;

    const int gw = blockIdx.x * WAVES_PB + wid;   // global wave id (0..8191)
    const int nw = N_BLOCKS * WAVES_PB;           // 8192 waves total

    float acc = 0.0f;

    // Prime the pipeline with the first row.
    issue_row_async(in + (size_t)gw * N_COLS, lds0, lane);

    int cur = 0;
    for (int it = 0; it < ROWS_PW; ++it) {
        const int row = gw + it * nw;

        if (it + 1 < ROWS_PW) {
            // Prefetch next row into the other buffer, then wait for only the
            // OLDER 32 async ops (in-order completion) -> copy/compute overlap.
            issue_row_async(in + (size_t)(row + nw) * N_COLS,
                            cur ? lds0 : lds1, lane);
            asm volatile("s_wait_asynccnt 0x20" ::: "memory");
        } else {
            asm volatile("s_wait_asynccnt 0x0" ::: "memory");
        }

        const float* slot = cur ? slot1 : slot0;

        // ---- pass 1: row max over logits (b128 LDS reads, tail masked)
        float m = -INFINITY;
        #pragma unroll
        for (int k = 0; k < 8; ++k) {
            const int base = k * 128 + lane * 4;
            const float4 v = *(const float4*)(slot + base);
            if (base + 0 < N_LOGITS) m = fmaxf(m, v.x);
            if (base + 1 < N_LOGITS) m = fmaxf(m, v.y);
            if (base + 2 < N_LOGITS) m = fmaxf(m, v.z);
            if (base + 3 < N_LOGITS) m = fmaxf(m, v.w);
        }
        m = wave_max(m);

        const float extra = slot[N_LOGITS];       // broadcast read

        // ---- pass 2: fused Z = sum(exp(l-m)), S = sum(exp(l-m)*softplus(l-extra))
        float Z = 0.0f, S = 0.0f;
        #pragma unroll
        for (int k = 0; k < 8; ++k) {
            const int base = k * 128 + lane * 4;
            const float4 v = *(const float4*)(slot + base);
            const float xs[4] = {v.x, v.y, v.z, v.w};
            #pragma unroll
            for (int j = 0; j < 4; ++j) {
                if (base + j < N_LOGITS) {
                    const float x  = xs[j];
                    const float e  = __expf(x - m);
                    const float t  = x - extra;
                    // softplus(t) = -log(sigmoid(extra - l)), stable form
                    const float sp = fmaxf(t, 0.0f)
                                   + __logf(1.0f + __expf(-fabsf(t)));
                    Z += e;
                    S += e * sp;
                }
            }
        }
        Z = wave_sum(Z);
        S = wave_sum(S);
        acc += S / Z;                              // wave-uniform after butterflies
        cur ^= 1;
    }

    if (lane == 0) wacc[wid] = acc;
    __syncthreads();
    if (wid == 0) {
        float v = (lane < WAVES_PB) ? wacc[lane] : 0.0f;
        v = wave_sum(v);
        if (lane == 0) partial[blockIdx.x] = v;
    }
}

__global__ __launch_bounds__(256)
void disc_adv_loss_reduce(const float* __restrict__ partial,
                          float* __restrict__ out, int n, float scale)
{
    __shared__ float s[256];
    float v = 0.0f;
    for (int i = threadIdx.x; i < n; i += 256) v += partial[i];
    s[threadIdx.x] = v;
    __syncthreads();
    for (int st = 128; st > 0; st >>= 1) {
        if ((int)threadIdx.x < st) s[threadIdx.x] += s[threadIdx.x + st];
        __syncthreads();
    }
    if (threadIdx.x == 0) out[0] = s[0] * scale;   // loss = (sum S/Z) / B
}

extern "C" void kernel_launch(void* const* d_in, const int* in_sizes, int n_in,
                              void* d_out, int out_size, void* d_ws, size_t ws_size,
                              hipStream_t stream)
{
    const float* in  = (const float*)d_in[0];   // (65536, 1001) f32
    // d_in[1] (target) is unused by the reference loss.
    float* out       = (float*)d_out;           // scalar f32
    float* partials  = (float*)d_ws;            // N_BLOCKS floats (4 KB)

    disc_adv_loss_rows<<<N_BLOCKS, THREADS_PB, 0, stream>>>(in, partials);
    disc_adv_loss_reduce<<<1, 256, 0, stream>>>(partials, out, N_BLOCKS,
                                                1.0f / (float)N_ROWS);
}